// AttGCNEncoder_77644418777421
// MI455X (gfx1250) — compile-verified
//
#include <hip/hip_runtime.h>

typedef __attribute__((ext_vector_type(2))) float v2f;
typedef __attribute__((ext_vector_type(8))) float v8f;

#define DFEAT 64

// ---------------- small helpers ----------------

__global__ void fill_i32(int* __restrict__ p, int v, long n) {
  long i = (long)blockIdx.x * blockDim.x + threadIdx.x;
  if (i < n) p[i] = v;
}

// in-degree count (excluding self-loop; that is added analytically later)
__global__ void count_accum(const long long* __restrict__ dst, int* __restrict__ count, long E) {
  long e = (long)blockIdx.x * blockDim.x + threadIdx.x;
  if (e < E) atomicAdd(&count[dst[e]], 1);
}

// dis[i] = rsqrt(deg) with deg = count + 1 (self-loop); deg >= 1 always
__global__ void inv_sqrt_k(const int* __restrict__ count, float* __restrict__ dis, long n) {
  long i = (long)blockIdx.x * blockDim.x + threadIdx.x;
  if (i < n) dis[i] = rsqrtf((float)count[i] + 1.0f);
}

// allocate a CSR segment per node: start[i] = fetch_add(total, count[i]); cursor starts there
__global__ void alloc_segments(const int* __restrict__ count, int* __restrict__ start,
                               int* __restrict__ cursor, int* __restrict__ total, long n) {
  long i = (long)blockIdx.x * blockDim.x + threadIdx.x;
  if (i < n) {
    int c = count[i];
    int s = atomicAdd(total, c);
    start[i]  = s;
    cursor[i] = s;
  }
}

// scatter edge sources into their destination's segment
__global__ void fill_edges(const long long* __restrict__ src, const long long* __restrict__ dst,
                           int* __restrict__ cursor, int* __restrict__ esrc, long E) {
  long e = (long)blockIdx.x * blockDim.x + threadIdx.x;
  if (e < E) {
    int pos = atomicAdd(&cursor[dst[e]], 1);
    esrc[pos] = (int)src[e];
  }
}

// ---------------- fused CSR aggregation + self-loop + bias + ReLU ----------------
// One wave32 per destination node; lane owns feature channels {2*lane, 2*lane+1}.
// out[d] = relu( dis[d] * sum_{s in nbrs(d)} hlin[s]*dis[s]  +  hlin[d]*dis[d]^2  +  b )

__global__ __launch_bounds__(256) void aggregate_csr(const int* __restrict__ esrc,
                                                     const int* __restrict__ start,
                                                     const int* __restrict__ count,
                                                     const float* __restrict__ hlin,
                                                     const float* __restrict__ dis,
                                                     const float* __restrict__ bias,
                                                     float* __restrict__ out, int nnodes) {
  const int lane = threadIdx.x & 31;
  const int node = blockIdx.x * 8 + (threadIdx.x >> 5);
  if (node >= nnodes) return;
  const int s0  = start[node];
  const int cnt = count[node];
  const float di = dis[node];

  float ax = 0.0f, ay = 0.0f;
  for (int p = 0; p < cnt; ++p) {
    int s = esrc[s0 + p];
    float w = dis[s];
    v2f m = ((const v2f*)(hlin + (long)s * DFEAT))[lane];
    ax += m.x * w;
    ay += m.y * w;
  }

  v2f hs = ((const v2f*)(hlin + (long)node * DFEAT))[lane];
  float v0 = ax * di + hs.x * di * di + bias[2 * lane + 0];
  float v1 = ay * di + hs.y * di * di + bias[2 * lane + 1];
  v2f o;
  o.x = v0 > 0.0f ? v0 : 0.0f;
  o.y = v1 > 0.0f ? v1 : 0.0f;
  ((v2f*)(out + (long)node * DFEAT))[lane] = o;
}

// ---------------- fp32 WMMA GEMM: Y[nrows x 64] = X[nrows x 64] @ W[64 x 64] (+ bias) ----
// One wave per 16-row tile; 4 column tiles of 16; K in steps of 4 via v_wmma_f32_16x16x4_f32.

__global__ __launch_bounds__(256) void gemm64_wmma(const float* __restrict__ X,
                                                   const float* __restrict__ W,
                                                   const float* __restrict__ bias,
                                                   float* __restrict__ Y, int nrows) {
  const int lane = threadIdx.x & 31;
  const int wave = threadIdx.x >> 5;
  const int half = lane >> 4;  // 0 or 1
  const int l16  = lane & 15;
  const long base = ((long)blockIdx.x * 8 + wave) * 16;
  if (base >= nrows) return;  // wave-uniform: EXEC stays all-ones for WMMA

  v8f acc[4] = {v8f{}, v8f{}, v8f{}, v8f{}};
  const float* __restrict__ xrow = X + (base + l16) * DFEAT;

#pragma unroll
  for (int k = 0; k < DFEAT; k += 4) {
    v2f a;
    a.x = xrow[k + 2 * half + 0];
    a.y = xrow[k + 2 * half + 1];
    const float* wr = W + (long)(k + 2 * half) * DFEAT + l16;
#pragma unroll
    for (int j = 0; j < 4; ++j) {
      v2f b;
      b.x = wr[j * 16];          // row K = k + 2*half
      b.y = wr[DFEAT + j * 16];  // row K = k + 2*half + 1
      acc[j] = __builtin_amdgcn_wmma_f32_16x16x4_f32(
          /*neg_a=*/false, a, /*neg_b=*/false, b,
          /*c_mod=*/(short)0, acc[j], /*reuse_a=*/false, /*reuse_b=*/false);
    }
  }

  float bb[4];
#pragma unroll
  for (int j = 0; j < 4; ++j) bb[j] = bias ? bias[j * 16 + l16] : 0.0f;

#pragma unroll
  for (int r = 0; r < 8; ++r) {
    long row = base + r + 8 * half;
    if (row < nrows) {
      float* yr = Y + row * DFEAT + l16;
      yr[0]  = acc[0][r] + bb[0];
      yr[16] = acc[1][r] + bb[1];
      yr[32] = acc[2][r] + bb[2];
      yr[48] = acc[3][r] + bb[3];
    }
  }
}

// ---------------- driver ----------------

static inline long round_up4(long x) { return (x + 3) & ~3L; }

extern "C" void kernel_launch(void* const* d_in, const int* in_sizes, int n_in,
                              void* d_out, int out_size, void* d_ws, size_t ws_size,
                              hipStream_t stream) {
  const float*     x  = (const float*)d_in[0];
  const long long* ei = (const long long*)d_in[1];
  const float*     W1 = (const float*)d_in[2];
  const float*     b1 = (const float*)d_in[3];
  const float*     W2 = (const float*)d_in[4];
  const float*     b2 = (const float*)d_in[5];
  // d_in[6..9] = Wq,bq,Wk,bk are dead: softmax over a length-1 axis is identically 1.
  const float*     Wv = (const float*)d_in[10];
  const float*     bv = (const float*)d_in[11];

  const long N  = in_sizes[0] / DFEAT;
  const long E  = in_sizes[1] / 2;
  const long ND = N * DFEAT;
  const long long* srcp = ei;
  const long long* dstp = ei + E;

  // workspace layout (4-byte elements, 16B-aligned sections)
  char* ws = (char*)d_ws;
  long off = 0;
  int*   count  = (int*)(ws + off); off += round_up4(N) * 4;
  int*   total  = (int*)(ws + off); off += round_up4(1) * 4;
  int*   start  = (int*)(ws + off); off += round_up4(N) * 4;
  int*   cursor = (int*)(ws + off); off += round_up4(N) * 4;
  int*   esrc   = (int*)(ws + off); off += round_up4(E) * 4;
  float* dis    = (float*)(ws + off); off += round_up4(N) * 4;
  float* bufA   = (float*)(ws + off); off += ND * 4;
  float* bufB   = (float*)(ws + off); off += ND * 4;

  dim3 blk(256);
  const int gN   = (int)((N + 255) / 256);
  const int gE   = (int)((E + 255) / 256);
  const int gNw  = (int)((N + 7) / 8);      // wave-per-node aggregation
  const int gGem = (int)((N + 127) / 128);  // 8 waves x 16 rows per block

  // ---- CSR build (once per launch): counts, normalization, segments, edge fill ----
  fill_i32<<<gN, blk, 0, stream>>>(count, 0, N);
  fill_i32<<<1, 32, 0, stream>>>(total, 0, 1);
  count_accum<<<gE, blk, 0, stream>>>(dstp, count, E);
  inv_sqrt_k<<<gN, blk, 0, stream>>>(count, dis, N);
  alloc_segments<<<gN, blk, 0, stream>>>(count, start, cursor, total, N);
  fill_edges<<<gE, blk, 0, stream>>>(srcp, dstp, cursor, esrc, E);

  // ---- layer 1: Hlin = X@W1 -> bufA; fused gather+selfloop+bias+relu -> bufB ----
  gemm64_wmma<<<gGem, blk, 0, stream>>>(x, W1, nullptr, bufA, (int)N);
  aggregate_csr<<<gNw, blk, 0, stream>>>(esrc, start, count, bufA, dis, b1, bufB, (int)N);

  // ---- layer 2: H2lin = H1@W2 -> bufA; fused aggregation -> bufB ----
  gemm64_wmma<<<gGem, blk, 0, stream>>>(bufB, W2, nullptr, bufA, (int)N);
  aggregate_csr<<<gNw, blk, 0, stream>>>(esrc, start, count, bufA, dis, b2, bufB, (int)N);

  // ---- output: context == v == H2@Wv + bv (softmax over singleton axis == 1) ----
  gemm64_wmma<<<gGem, blk, 0, stream>>>(bufB, Wv, bv, (float*)d_out, (int)N);
}